// DeeperGCN_90297392431233
// MI455X (gfx1250) — compile-verified
//
#include <hip/hip_runtime.h>
#include <cstdint>

// ---------------------------------------------------------------------------
// DeeperGCN on gfx1250 (MI455X). f32 WMMA (16x16x4) for all GEMMs, L2-resident
// atomic scatter for softmax aggregation, fused LN/ReLU/residual in the MLP.
// ---------------------------------------------------------------------------

typedef __attribute__((ext_vector_type(2))) float v2f;
typedef __attribute__((ext_vector_type(8))) float v8f;

#define WMMA_F32(a, b, c) \
  __builtin_amdgcn_wmma_f32_16x16x4_f32(false, (a), false, (b), (short)0, (c), false, false)

constexpr int Hdim = 64;    // feature dim
constexpr int H2   = 128;   // hidden dim
constexpr int IND  = 128;   // input dim

__device__ __forceinline__ void atomicMaxFloat(float* addr, float v) {
  // monotonic split trick: positives via signed max, negatives via unsigned min
  if (v >= 0.f) atomicMax((int*)addr, __float_as_int(v));
  else          atomicMin((unsigned int*)addr, __float_as_uint(v));
}

// ---------------------------------------------------------------------------
// Encoder: h = x @ enc_W + enc_b      x:[N,128] W:[128,64]
// 64 rows per block, A tile in LDS, W streamed from L2.
// ---------------------------------------------------------------------------
__global__ __launch_bounds__(256) void enc_gemm_kernel(
    const float* __restrict__ x, const float* __restrict__ W,
    const float* __restrict__ bias, float* __restrict__ h, int nrows)
{
  __shared__ float Xs[64][IND + 4];
  const int tid  = threadIdx.x;
  const int row0 = blockIdx.x * 64;

  for (int i = tid; i < 64 * (IND / 4); i += 256) {
    int r  = i / (IND / 4);
    int c4 = (i % (IND / 4)) * 4;
    int gr = row0 + r;
    float4 v = make_float4(0.f, 0.f, 0.f, 0.f);
    if (gr < nrows) v = *(const float4*)(x + (long long)gr * IND + c4);
    Xs[r][c4 + 0] = v.x; Xs[r][c4 + 1] = v.y;
    Xs[r][c4 + 2] = v.z; Xs[r][c4 + 3] = v.w;
  }
  __syncthreads();

  const int wave = tid >> 5, lane = tid & 31;
  const int half = lane >> 4, ln = lane & 15;

  for (int t = 0; t < 2; ++t) {               // 16 tiles (4x4) over 8 waves
    int tile = wave * 2 + t;
    int m0 = (tile >> 2) * 16, n0 = (tile & 3) * 16;
    v8f acc = {};
    for (int kk = 0; kk < IND; kk += 4) {
      int ka = kk + 2 * half;
      v2f a, b;
      a[0] = Xs[m0 + ln][ka];      a[1] = Xs[m0 + ln][ka + 1];
      b[0] = W[ka * Hdim + n0 + ln];
      b[1] = W[(ka + 1) * Hdim + n0 + ln];
      acc = WMMA_F32(a, b, acc);
    }
    for (int r = 0; r < 8; ++r) {
      int gr = row0 + m0 + r + 8 * half;
      if (gr < nrows) {
        int nn = n0 + ln;
        h[(long long)gr * Hdim + nn] = acc[r] + bias[nn];
      }
    }
  }
}

// ---------------------------------------------------------------------------
// z = relu(layer_norm(h, g, b))   one wave (32 lanes) per 64-wide row
// ---------------------------------------------------------------------------
__global__ __launch_bounds__(256) void ln_relu_kernel(
    const float* __restrict__ h, const float* __restrict__ g,
    const float* __restrict__ b, float* __restrict__ z, int nrows)
{
  int row  = blockIdx.x * 8 + (threadIdx.x >> 5);
  int lane = threadIdx.x & 31;
  if (row >= nrows) return;
  const float* hr = h + (long long)row * Hdim;
  float v0 = hr[lane], v1 = hr[lane + 32];
  float s = v0 + v1, s2 = v0 * v0 + v1 * v1;
  for (int off = 1; off < 32; off <<= 1) {
    s  += __shfl_xor(s, off);
    s2 += __shfl_xor(s2, off);
  }
  float mu  = s * (1.f / 64.f);
  float var = s2 * (1.f / 64.f) - mu * mu;
  float rs  = rsqrtf(var + 1e-5f);
  float* zr = z + (long long)row * Hdim;
  zr[lane]      = fmaxf((v0 - mu) * rs * g[lane]      + b[lane],      0.f);
  zr[lane + 32] = fmaxf((v1 - mu) * rs * g[lane + 32] + b[lane + 32], 0.f);
}

// ---------------------------------------------------------------------------
// init aggregation buffers
// ---------------------------------------------------------------------------
__global__ void init_aggr_kernel(float* __restrict__ mmax, float* __restrict__ denom,
                                 float* __restrict__ numer, long long total)
{
  long long i = (long long)blockIdx.x * blockDim.x + threadIdx.x;
  if (i < total) {
    mmax[i]  = -3.402823466e38f;
    denom[i] = 0.f;
    numer[i] = 0.f;
  }
}

// ---------------------------------------------------------------------------
// Edge pass 1: segment-max of logits = (relu(z[src]) + eps) * t   (L2 atomics)
// one thread handles one edge x 4 features (float4 gather, 16 threads/edge)
// ---------------------------------------------------------------------------
__global__ __launch_bounds__(256) void edge_max_kernel(
    const float* __restrict__ z, const int* __restrict__ src,
    const int* __restrict__ dst, const float* __restrict__ t, int ti,
    float* __restrict__ mmax, int E)
{
  long long tid = (long long)blockIdx.x * blockDim.x + threadIdx.x;
  if (tid >= (long long)E * 16) return;
  int e = (int)(tid >> 4);
  int f = ((int)tid & 15) * 4;
  int s = src[e], d = dst[e];
  float tv = t[ti];
  float4 zv = *(const float4*)(z + (long long)s * Hdim + f);
  float* mp = mmax + (long long)d * Hdim + f;
  atomicMaxFloat(mp + 0, (fmaxf(zv.x, 0.f) + 1e-7f) * tv);
  atomicMaxFloat(mp + 1, (fmaxf(zv.y, 0.f) + 1e-7f) * tv);
  atomicMaxFloat(mp + 2, (fmaxf(zv.z, 0.f) + 1e-7f) * tv);
  atomicMaxFloat(mp + 3, (fmaxf(zv.w, 0.f) + 1e-7f) * tv);
}

// ---------------------------------------------------------------------------
// Edge pass 2: alpha = exp(logit - mmax[dst]); denom += alpha; numer += msg*alpha
// ---------------------------------------------------------------------------
__global__ __launch_bounds__(256) void edge_alpha_kernel(
    const float* __restrict__ z, const int* __restrict__ src,
    const int* __restrict__ dst, const float* __restrict__ t, int ti,
    const float* __restrict__ mmax, float* __restrict__ denom,
    float* __restrict__ numer, int E)
{
  long long tid = (long long)blockIdx.x * blockDim.x + threadIdx.x;
  if (tid >= (long long)E * 16) return;
  int e = (int)(tid >> 4);
  int f = ((int)tid & 15) * 4;
  int s = src[e], d = dst[e];
  float tv = t[ti];
  float4 zv = *(const float4*)(z + (long long)s * Hdim + f);
  float4 mv = *(const float4*)(mmax + (long long)d * Hdim + f);
  float* dp = denom + (long long)d * Hdim + f;
  float* np = numer + (long long)d * Hdim + f;

  float m0 = fmaxf(zv.x, 0.f) + 1e-7f; float a0 = __expf(m0 * tv - mv.x);
  float m1 = fmaxf(zv.y, 0.f) + 1e-7f; float a1 = __expf(m1 * tv - mv.y);
  float m2 = fmaxf(zv.z, 0.f) + 1e-7f; float a2 = __expf(m2 * tv - mv.z);
  float m3 = fmaxf(zv.w, 0.f) + 1e-7f; float a3 = __expf(m3 * tv - mv.w);
  atomicAdd(dp + 0, a0); atomicAdd(np + 0, m0 * a0);
  atomicAdd(dp + 1, a1); atomicAdd(np + 1, m1 * a1);
  atomicAdd(dp + 2, a2); atomicAdd(np + 2, m2 * a2);
  atomicAdd(dp + 3, a3); atomicAdd(np + 3, m3 * a3);
}

// ---------------------------------------------------------------------------
// Fused conv MLP per 64-row tile:
//   u = z + numer/(denom+eps)            (aggregation epilogue fused in)
//   t1 = relu(LN(u @ W1 + b1))           GEMM1 + LN in LDS
//   h_out = [base +] t1 @ W2 + b2        GEMM2 + residual
// ---------------------------------------------------------------------------
__global__ __launch_bounds__(256) void mlp_kernel(
    const float* __restrict__ z, const float* __restrict__ numer,
    const float* __restrict__ denom, const float* __restrict__ base,
    const float* __restrict__ W1, const float* __restrict__ b1,
    const float* __restrict__ g1, const float* __restrict__ be1,
    const float* __restrict__ W2, const float* __restrict__ b2,
    float* __restrict__ hout, int nrows)
{
  __shared__ float U[64][Hdim + 4];   // u tile
  __shared__ float T[64][H2 + 4];     // hidden tile

  const int tid  = threadIdx.x;
  const int row0 = blockIdx.x * 64;

  // u = z + numer / (denom + eps)
  for (int i = tid; i < 64 * Hdim; i += 256) {
    int r = i >> 6, c = i & 63;
    int gr = row0 + r;
    float v = 0.f;
    if (gr < nrows) {
      long long idx = (long long)gr * Hdim + c;
      v = z[idx] + numer[idx] / (denom[idx] + 1e-16f);
    }
    U[r][c] = v;
  }
  __syncthreads();

  const int wave = tid >> 5, lane = tid & 31;
  const int half = lane >> 4, ln = lane & 15;

  // GEMM1: T = U[64x64] @ W1[64x128]   (32 tiles over 8 waves)
  for (int t = 0; t < 4; ++t) {
    int tile = wave * 4 + t;
    int m0 = (tile >> 3) * 16, n0 = (tile & 7) * 16;
    v8f acc = {};
    for (int kk = 0; kk < Hdim; kk += 4) {
      int ka = kk + 2 * half;
      v2f a, b;
      a[0] = U[m0 + ln][ka];     a[1] = U[m0 + ln][ka + 1];
      b[0] = W1[ka * H2 + n0 + ln];
      b[1] = W1[(ka + 1) * H2 + n0 + ln];
      acc = WMMA_F32(a, b, acc);
    }
    for (int r = 0; r < 8; ++r) {
      int mm = m0 + r + 8 * half;
      int nn = n0 + ln;
      T[mm][nn] = acc[r] + b1[nn];
    }
  }
  __syncthreads();

  // LayerNorm + ReLU over 128-wide rows; 4 lanes per row
  {
    int r = tid >> 2;          // 0..63
    int q = tid & 3;           // 0..3
    float s = 0.f, s2 = 0.f;
    for (int c = q * 32; c < q * 32 + 32; ++c) {
      float v = T[r][c]; s += v; s2 += v * v;
    }
    s  += __shfl_xor(s, 1);  s  += __shfl_xor(s, 2);
    s2 += __shfl_xor(s2, 1); s2 += __shfl_xor(s2, 2);
    float mu  = s * (1.f / 128.f);
    float var = s2 * (1.f / 128.f) - mu * mu;
    float rs  = rsqrtf(var + 1e-5f);
    for (int c = q * 32; c < q * 32 + 32; ++c) {
      float v = (T[r][c] - mu) * rs * g1[c] + be1[c];
      T[r][c] = fmaxf(v, 0.f);
    }
  }
  __syncthreads();

  // GEMM2: out = T[64x128] @ W2[128x64]   (16 tiles over 8 waves)
  for (int t = 0; t < 2; ++t) {
    int tile = wave * 2 + t;
    int m0 = (tile >> 2) * 16, n0 = (tile & 3) * 16;
    v8f acc = {};
    for (int kk = 0; kk < H2; kk += 4) {
      int ka = kk + 2 * half;
      v2f a, b;
      a[0] = T[m0 + ln][ka];     a[1] = T[m0 + ln][ka + 1];
      b[0] = W2[ka * Hdim + n0 + ln];
      b[1] = W2[(ka + 1) * Hdim + n0 + ln];
      acc = WMMA_F32(a, b, acc);
    }
    for (int r = 0; r < 8; ++r) {
      int gr = row0 + m0 + r + 8 * half;
      if (gr < nrows) {
        int nn = n0 + ln;
        long long idx = (long long)gr * Hdim + nn;
        float v = acc[r] + b2[nn];
        if (base) v += base[idx];
        hout[idx] = v;
      }
    }
  }
}

// ---------------------------------------------------------------------------
// Final: h_fin = relu(LN(h, ng0, nb0)) -> out[:N*64]; logits = h_fin@linW+linb
// ---------------------------------------------------------------------------
__global__ __launch_bounds__(256) void final_kernel(
    const float* __restrict__ h, const float* __restrict__ g,
    const float* __restrict__ b, const float* __restrict__ linW,
    const float* __restrict__ linB, float* __restrict__ outH,
    float* __restrict__ outL, int nrows)
{
  int row  = blockIdx.x * 8 + (threadIdx.x >> 5);
  int lane = threadIdx.x & 31;
  if (row >= nrows) return;
  const float* hr = h + (long long)row * Hdim;
  float v0 = hr[lane], v1 = hr[lane + 32];
  float s = v0 + v1, s2 = v0 * v0 + v1 * v1;
  for (int off = 1; off < 32; off <<= 1) {
    s  += __shfl_xor(s, off);
    s2 += __shfl_xor(s2, off);
  }
  float mu  = s * (1.f / 64.f);
  float var = s2 * (1.f / 64.f) - mu * mu;
  float rs  = rsqrtf(var + 1e-5f);
  float f0 = fmaxf((v0 - mu) * rs * g[lane]      + b[lane],      0.f);
  float f1 = fmaxf((v1 - mu) * rs * g[lane + 32] + b[lane + 32], 0.f);
  outH[(long long)row * Hdim + lane]      = f0;
  outH[(long long)row * Hdim + lane + 32] = f1;
  for (int j = 0; j < 3; ++j) {
    float p = f0 * linW[lane * 3 + j] + f1 * linW[(lane + 32) * 3 + j];
    for (int off = 1; off < 32; off <<= 1) p += __shfl_xor(p, off);
    if (lane == 0) outL[(long long)row * 3 + j] = p + linB[j];
  }
}

// ---------------------------------------------------------------------------
extern "C" void kernel_launch(void* const* d_in, const int* in_sizes, int n_in,
                              void* d_out, int out_size, void* d_ws, size_t ws_size,
                              hipStream_t stream)
{
  const float* x    = (const float*)d_in[0];
  const int*   edge = (const int*)d_in[1];
  const float* encW = (const float*)d_in[2];
  const float* encB = (const float*)d_in[3];
  const float* tvec = (const float*)d_in[4];
  const float* W1   = (const float*)d_in[5];
  const float* b1   = (const float*)d_in[6];
  const float* g1   = (const float*)d_in[7];
  const float* be1  = (const float*)d_in[8];
  const float* W2   = (const float*)d_in[9];
  const float* b2   = (const float*)d_in[10];
  const float* ng   = (const float*)d_in[11];
  const float* nb   = (const float*)d_in[12];
  const float* linW = (const float*)d_in[13];
  const float* linB = (const float*)d_in[14];

  const int N = in_sizes[0] / IND;
  const int E = in_sizes[1] / 2;
  const int* src = edge;
  const int* dst = edge + E;

  float* h     = (float*)d_ws;
  float* z     = h     + (size_t)N * Hdim;
  float* mmax  = z     + (size_t)N * Hdim;
  float* denom = mmax  + (size_t)N * Hdim;
  float* numer = denom + (size_t)N * Hdim;

  const int rowBlocks  = (N + 63) / 64;
  const int waveBlocks = (N + 7) / 8;
  const long long nf   = (long long)N * Hdim;
  const int nfBlocks   = (int)((nf + 255) / 256);
  const long long et   = (long long)E * 16;
  const int eBlocks    = (int)((et + 255) / 256);

  enc_gemm_kernel<<<rowBlocks, 256, 0, stream>>>(x, encW, encB, h, N);

  for (int i = 0; i < 3; ++i) {
    const float* zin;
    const float* base;
    if (i == 0) {
      zin = h; base = nullptr;                 // layer 0: raw conv on h
    } else {
      ln_relu_kernel<<<waveBlocks, 256, 0, stream>>>(h, ng + i * Hdim, nb + i * Hdim, z, N);
      zin = z; base = h;                       // res+ block
    }
    init_aggr_kernel<<<nfBlocks, 256, 0, stream>>>(mmax, denom, numer, nf);
    edge_max_kernel<<<eBlocks, 256, 0, stream>>>(zin, src, dst, tvec, i, mmax, E);
    edge_alpha_kernel<<<eBlocks, 256, 0, stream>>>(zin, src, dst, tvec, i, mmax, denom, numer, E);
    mlp_kernel<<<rowBlocks, 256, 0, stream>>>(
        zin, numer, denom, base,
        W1 + (size_t)i * Hdim * H2, b1 + (size_t)i * H2,
        g1 + (size_t)i * H2,        be1 + (size_t)i * H2,
        W2 + (size_t)i * H2 * Hdim, b2 + (size_t)i * Hdim, h, N);
  }

  float* outH = (float*)d_out;
  float* outL = outH + (size_t)N * Hdim;
  final_kernel<<<waveBlocks, 256, 0, stream>>>(h, ng, nb, linW, linB, outH, outL, N);
}